// SSelfAttention_45217415692336
// MI455X (gfx1250) — compile-verified
//
#include <hip/hip_runtime.h>

typedef float v2f __attribute__((ext_vector_type(2)));
typedef float v8f __attribute__((ext_vector_type(8)));

#define EMBED 64
#define HEADS 8
#define HDIM  8
#define BB    8
#define NN    128
#define TT    64

__device__ __forceinline__ v8f wmma_f32_16x16x4(v2f a, v2f b, v8f c) {
  // 8 args: (neg_a, A, neg_b, B, c_mod, C, reuse_a, reuse_b)
  return __builtin_amdgcn_wmma_f32_16x16x4_f32(false, a, false, b, (short)0, c,
                                               false, false);
}

// One block = one (b, t, h, rowHalf): computes 64 query rows of the fused
// projection -> QK^T -> softmax -> @V attention, entirely in LDS.
__global__ __launch_bounds__(128) void attn_kernel(
    const float* __restrict__ V, const float* __restrict__ K,
    const float* __restrict__ Q, const float* __restrict__ Wv,
    const float* __restrict__ Wk, const float* __restrict__ Wq,
    float* __restrict__ Xout) {
  __shared__ float S[64 * NN];       // 32 KB: energy/prob tile (64 q-rows x 128 keys)
  __shared__ float Qp[NN * HDIM];    // projected Q (all 128 rows; cheap)
  __shared__ float Kp[NN * HDIM];
  __shared__ float Vp[NN * HDIM];
  __shared__ float redM[128];        // softmax partial max (2 per row)
  __shared__ float redS[128];        // softmax partial sum (2 per row)

  const int blk     = blockIdx.x;            // (((b*T)+t)*H + h)*2 + rh
  const int rh      = blk & 1;
  const int h       = (blk >> 1) % HEADS;
  const int t       = (blk >> 1) / HEADS % TT;
  const int b       = (blk >> 1) / (HEADS * TT);
  const int rowBase = rh * 64;               // query-row offset within N

  const int tid  = threadIdx.x;              // 0..127
  const int wave = tid >> 5;                 // 0..3
  const int lane = tid & 31;
  const int half = lane >> 4;                // lane group (K/M split per ISA layout)
  const int l    = lane & 15;

  // ---- Phase 0: load q/k/v rows for this (b,t,h), apply 8x8 projections ----
  {
    const int n = tid;
    const size_t base = (((size_t)(b * NN + n) * TT + t) * EMBED) + h * HDIM;
    float qv[HDIM], kv[HDIM], vv[HDIM];
#pragma unroll
    for (int d = 0; d < HDIM; ++d) {
      qv[d] = Q[base + d];
      kv[d] = K[base + d];
      vv[d] = V[base + d];
    }
#pragma unroll
    for (int e = 0; e < HDIM; ++e) {
      float aq = 0.f, ak = 0.f, av = 0.f;
#pragma unroll
      for (int d = 0; d < HDIM; ++d) {
        aq += qv[d] * Wq[e * HDIM + d];
        ak += kv[d] * Wk[e * HDIM + d];
        av += vv[d] * Wv[e * HDIM + d];
      }
      Qp[n * HDIM + e] = aq;
      Kp[n * HDIM + e] = ak;
      Vp[n * HDIM + e] = av;
    }
  }
  __syncthreads();

  // ---- Phase 1: S = (Qp @ Kp^T) * 1/sqrt(EMBED); 4x8 grid of 16x16 tiles ----
  const float scale = 0.125f;  // 1/sqrt(64)
  for (int tile = wave; tile < 32; tile += 4) {
    const int ti = tile >> 3;  // 0..3 (local q-row tile)
    const int tj = tile & 7;   // 0..7 (key tile)
    v8f c = {};
#pragma unroll
    for (int kb = 0; kb < HDIM; kb += 4) {
      const int ac = kb + 2 * half;  // K index held by this lane group
      v2f a, bm;
      a.x  = Qp[(rowBase + ti * 16 + l) * HDIM + ac];
      a.y  = Qp[(rowBase + ti * 16 + l) * HDIM + ac + 1];
      bm.x = Kp[(tj * 16 + l) * HDIM + ac];   // B = Kp^T -> B[k][n] = Kp[n][k]
      bm.y = Kp[(tj * 16 + l) * HDIM + ac + 1];
      c = wmma_f32_16x16x4(a, bm, c);
    }
#pragma unroll
    for (int r = 0; r < 8; ++r) {
      const int row = ti * 16 + r + 8 * half;  // C/D layout: VGPR r, M = r + 8*half
      S[row * NN + tj * 16 + l] = c[r] * scale;
    }
  }
  __syncthreads();

  // ---- Phase 2: row softmax over 128 keys; 2 threads per row (64 cols each) --
  {
    const int row = tid >> 1;
    const int seg = tid & 1;
    float* Sseg = S + row * NN + seg * 64;

    float m = -1e30f;
    for (int j = 0; j < 64; ++j) m = fmaxf(m, Sseg[j]);
    redM[tid] = m;
    __syncthreads();
    m = fmaxf(redM[row * 2], redM[row * 2 + 1]);

    float sum = 0.f;
    for (int j = 0; j < 64; ++j) {
      const float ex = __expf(Sseg[j] - m);
      Sseg[j] = ex;
      sum += ex;
    }
    redS[tid] = sum;
    __syncthreads();
    const float inv = 1.f / (redS[row * 2] + redS[row * 2 + 1]);
    for (int j = 0; j < 64; ++j) Sseg[j] *= inv;
  }
  __syncthreads();

  // ---- Phase 3: O = P @ Vp  (M=64, N=8 padded to 16, K=128) ----
  {
    const int ti = wave;  // one 16-row tile per wave
    v8f c = {};
    for (int kb = 0; kb < NN; kb += 4) {
      const int ac = kb + 2 * half;
      v2f a, bm;
      a.x  = S[(ti * 16 + l) * NN + ac];
      a.y  = S[(ti * 16 + l) * NN + ac + 1];
      // B = Vp (128 x 8), pad columns 8..15 with zeros
      bm.x = (l < HDIM) ? Vp[ac * HDIM + l] : 0.f;
      bm.y = (l < HDIM) ? Vp[(ac + 1) * HDIM + l] : 0.f;
      c = wmma_f32_16x16x4(a, bm, c);
    }
    if (l < HDIM) {
#pragma unroll
      for (int r = 0; r < 8; ++r) {
        const int n = rowBase + ti * 16 + r + 8 * half;
        const size_t o = (((size_t)(b * NN + n) * TT + t) * EMBED) + h * HDIM + l;
        Xout[o] = c[r];
      }
    }
  }
}

// Y = X @ Wo^T + bo over 65536 rows. One block = one 16-row M-tile; 4 waves
// cover the 4 N-tiles of the 64-wide output.
__global__ __launch_bounds__(128) void fc_kernel(const float* __restrict__ X,
                                                 const float* __restrict__ Wo,
                                                 const float* __restrict__ bo,
                                                 float* __restrict__ Y) {
  __shared__ float sX[16 * EMBED];      // 4 KB
  __shared__ float sW[EMBED * EMBED];   // 16 KB
  __shared__ float sb[EMBED];

  const int tid  = threadIdx.x;
  const int wave = tid >> 5;
  const int lane = tid & 31;
  const int half = lane >> 4;
  const int l    = lane & 15;
  const size_t mBase = (size_t)blockIdx.x * 16;

  for (int i = tid; i < 16 * EMBED; i += 128) sX[i] = X[mBase * EMBED + i];
  for (int i = tid; i < EMBED * EMBED; i += 128) sW[i] = Wo[i];
  if (tid < EMBED) sb[tid] = bo[tid];
  __syncthreads();

  const int tj = wave;  // output-column tile 0..3
  v8f c = {};
#pragma unroll
  for (int kb = 0; kb < EMBED; kb += 4) {
    const int ac = kb + 2 * half;
    v2f a, bm;
    a.x  = sX[l * EMBED + ac];
    a.y  = sX[l * EMBED + ac + 1];
    bm.x = sW[(tj * 16 + l) * EMBED + ac];  // B = Wo^T -> B[k][n] = Wo[n][k]
    bm.y = sW[(tj * 16 + l) * EMBED + ac + 1];
    c = wmma_f32_16x16x4(a, bm, c);
  }
  const float bias = sb[tj * 16 + l];
#pragma unroll
  for (int r = 0; r < 8; ++r) {
    const size_t row = mBase + r + 8 * half;
    Y[row * EMBED + tj * 16 + l] = c[r] + bias;
  }
}

extern "C" void kernel_launch(void* const* d_in, const int* in_sizes, int n_in,
                              void* d_out, int out_size, void* d_ws,
                              size_t ws_size, hipStream_t stream) {
  (void)in_sizes; (void)n_in; (void)out_size; (void)ws_size;
  const float* V  = (const float*)d_in[0];
  const float* K  = (const float*)d_in[1];
  const float* Q  = (const float*)d_in[2];
  const float* Wv = (const float*)d_in[3];
  const float* Wk = (const float*)d_in[4];
  const float* Wq = (const float*)d_in[5];
  const float* Wo = (const float*)d_in[6];
  const float* bo = (const float*)d_in[7];
  float* X = (float*)d_ws;    // (B,N,T,EMBED) head-concat intermediate, 16 MB
  float* Y = (float*)d_out;

  hipLaunchKernelGGL(attn_kernel, dim3(BB * TT * HEADS * 2), dim3(128), 0,
                     stream, V, K, Q, Wv, Wk, Wq, X);
  hipLaunchKernelGGL(fc_kernel, dim3((BB * NN * TT) / 16), dim3(128), 0, stream,
                     X, Wo, bo, Y);
}